// NGCF_60035052863932
// MI455X (gfx1250) — compile-verified
//
#include <hip/hip_runtime.h>

#define N_NODES 200000
#define DLAT 64
#define OUT_STRIDE 256
#define NEG 0.2f

typedef __attribute__((ext_vector_type(2))) float v2f;
typedef __attribute__((ext_vector_type(8))) float v8f;

// out[:, 0:64] = raw input embedding
__global__ __launch_bounds__(256) void copy_embed_kernel(const float* __restrict__ embed,
                                                         float* __restrict__ out) {
    int tid = blockIdx.x * 256 + threadIdx.x;
    if (tid < N_NODES * DLAT) {
        int row = tid >> 6, col = tid & 63;
        out[(size_t)row * OUT_STRIDE + col] = embed[tid];
    }
}

// nei[dst] += val * x[src]   (COO scatter; 32 lanes per edge, float2 per lane)
__global__ __launch_bounds__(256) void spmm_kernel(const float* __restrict__ x,
                                                   const float* __restrict__ val,
                                                   const int* __restrict__ src,
                                                   const int* __restrict__ dst,
                                                   float* __restrict__ nei, int E) {
    long long t = (long long)blockIdx.x * 256 + threadIdx.x;
    int e = (int)(t >> 5);
    int lane = (int)(t & 31);
    if (e < E) {
        int s = src[e], d = dst[e];
        float v = val[e];
        float2 xv = *((const float2*)(x + (size_t)s * DLAT) + lane);
        float* np = nei + (size_t)d * DLAT + lane * 2;
        unsafeAtomicAdd(np,     v * xv.x);   // global_atomic_add_f32
        unsafeAtomicAdd(np + 1, v * xv.y);
    }
}

// Fused: Y = lrelu((nei+cur)@(W1+b1)) + lrelu((nei*cur)@(W2+b2));
// next (= nei, in place) <- Y raw;  out slice <- Y / max(||Y||, eps)
//
// Weights staged in LDS K-pair interleaved: (k,c) -> (k>>1)*128 + 2*c + (k&1),
// so a B-fragment {W'[kb][c], W'[kb+1][c]} (kb even) is one aligned ds_load_b64.
__global__ __launch_bounds__(256) void gemm_layer_kernel(
        const float* nei, const float* cur,
        const float* __restrict__ W1, const float* __restrict__ b1,
        const float* __restrict__ W2, const float* __restrict__ b2,
        float* next, float* __restrict__ out, int out_col) {
    __shared__ float sW1[DLAT * DLAT];
    __shared__ float sW2[DLAT * DLAT];
    for (int i = threadIdx.x; i < DLAT * DLAT; i += 256) {
        int k = i >> 6, c = i & 63;
        int idx = ((k >> 1) << 7) + (c << 1) + (k & 1);
        sW1[idx] = W1[i] + b1[c];   // bias broadcast over weight rows (torch quirk)
        sW2[idx] = W2[i] + b2[c];
    }
    __syncthreads();

    const int lane = threadIdx.x & 31;
    const int wave = threadIdx.x >> 5;
    const int half = lane >> 4;      // which 16-lane half
    const int l16  = lane & 15;

    const int ntiles = N_NODES / 16;
    int tile = blockIdx.x * 8 + wave;          // wave-uniform guard -> EXEC all 1s
    if (tile >= ntiles) return;
    const int base = tile * 16;

    v8f z = {0.f, 0.f, 0.f, 0.f, 0.f, 0.f, 0.f, 0.f};
    v8f acc1[4] = {z, z, z, z};
    v8f acc2[4] = {z, z, z, z};

    const float* neirow = nei + (size_t)(base + l16) * DLAT;
    const float* currow = cur + (size_t)(base + l16) * DLAT;
    // per-lane LDS base: half selects the odd K-pair, l16*2 selects the column pair
    const int wbase = half * 128 + l16 * 2;

#pragma unroll
    for (int k0 = 0; k0 < DLAT; k0 += 4) {
        int kb = k0 + 2 * half;                // A layout: K = k0 + 2*(lane/16) + j
        float2 nv = *(const float2*)(neirow + kb);
        float2 cv = *(const float2*)(currow + kb);
        v2f a1, a2;
        a1.x = nv.x + cv.x;  a1.y = nv.y + cv.y;   // sum branch
        a2.x = nv.x * cv.x;  a2.y = nv.y * cv.y;   // bi (hadamard) branch
#pragma unroll
        for (int nt = 0; nt < 4; ++nt) {
            int woff = wbase + (k0 >> 1) * 128 + nt * 32;  // folds into DS imm offset
            v2f bf1 = *(const v2f*)(sW1 + woff);
            v2f bf2 = *(const v2f*)(sW2 + woff);
            acc1[nt] = __builtin_amdgcn_wmma_f32_16x16x4_f32(
                false, a1, false, bf1, (short)0, acc1[nt], false, false);
            acc2[nt] = __builtin_amdgcn_wmma_f32_16x16x4_f32(
                false, a2, false, bf2, (short)0, acc2[nt], false, false);
        }
    }

    // Epilogue: leaky-relu, sum, row L2 norm (row lives in one 16-lane half)
    float yreg[4][8];
    float rowsq[8];
#pragma unroll
    for (int v = 0; v < 8; ++v) rowsq[v] = 0.f;
#pragma unroll
    for (int nt = 0; nt < 4; ++nt) {
#pragma unroll
        for (int v = 0; v < 8; ++v) {
            float y1 = acc1[nt][v]; y1 = fmaxf(y1, NEG * y1);
            float y2 = acc2[nt][v]; y2 = fmaxf(y2, NEG * y2);
            float y = y1 + y2;
            yreg[nt][v] = y;
            rowsq[v] += y * y;
        }
    }
#pragma unroll
    for (int v = 0; v < 8; ++v) {
        float s = rowsq[v];
        s += __shfl_xor(s, 1, 16);
        s += __shfl_xor(s, 2, 16);
        s += __shfl_xor(s, 4, 16);
        s += __shfl_xor(s, 8, 16);
        rowsq[v] = 1.0f / fmaxf(sqrtf(s), 1e-12f);  // 1/max(||row||, eps)
    }
#pragma unroll
    for (int nt = 0; nt < 4; ++nt) {
#pragma unroll
        for (int v = 0; v < 8; ++v) {
            int row = base + v + 8 * half;      // C/D layout: M = v + 8*(lane/16)
            int col = nt * 16 + l16;
            float y = yreg[nt][v];
            next[(size_t)row * DLAT + col] = y;                       // raw, feeds next layer
            out[(size_t)row * OUT_STRIDE + out_col + col] = y * rowsq[v];  // normalized slice
        }
    }
}

extern "C" void kernel_launch(void* const* d_in, const int* in_sizes, int n_in,
                              void* d_out, int out_size, void* d_ws, size_t ws_size,
                              hipStream_t stream) {
    (void)n_in; (void)out_size; (void)ws_size;
    const float* embed    = (const float*)d_in[0];
    const float* W1       = (const float*)d_in[1];
    const float* b1       = (const float*)d_in[2];
    const float* W2       = (const float*)d_in[3];
    const float* b2       = (const float*)d_in[4];
    const float* edge_val = (const float*)d_in[5];
    const int*   edge_src = (const int*)d_in[6];
    const int*   edge_dst = (const int*)d_in[7];
    float* out = (float*)d_out;
    const int E = in_sizes[5];

    const size_t bufElems = (size_t)N_NODES * DLAT;
    float* bufs[2] = { (float*)d_ws, (float*)d_ws + bufElems };

    copy_embed_kernel<<<(N_NODES * DLAT + 255) / 256, 256, 0, stream>>>(embed, out);

    const float* cur = embed;
    for (int k = 0; k < 3; ++k) {
        float* nei = bufs[k & 1];               // ping-pong; GEMM writes Y in-place over nei
        hipMemsetAsync(nei, 0, bufElems * sizeof(float), stream);
        long long tthreads = (long long)E * 32;
        int spmmBlocks = (int)((tthreads + 255) / 256);
        spmm_kernel<<<spmmBlocks, 256, 0, stream>>>(cur, edge_val, edge_src, edge_dst, nei, E);
        int gblocks = (N_NODES / 16 + 7) / 8;
        gemm_layer_kernel<<<gblocks, 256, 0, stream>>>(
            nei, cur,
            W1 + (size_t)k * DLAT * DLAT, b1 + (size_t)k * DLAT,
            W2 + (size_t)k * DLAT * DLAT, b2 + (size_t)k * DLAT,
            nei, out, 64 * (k + 1));
        cur = nei;
    }
}